// SiReN_75161927680657
// MI455X (gfx1250) — compile-verified
//
#include <hip/hip_runtime.h>
#include <hip/hip_bf16.h>
#include <math.h>

typedef __attribute__((ext_vector_type(2))) float v2f;
typedef __attribute__((ext_vector_type(8))) float v8f;

#define NNODES 150000
#define NDIM   64
#define NB     4096
#define NK     40
#define REGC   1e-4f
#define LW     68   // padded LDS row stride (bank-conflict free for stride-64 data)

// ---------------------------------------------------------------- utilities
__global__ void k_zero(float* __restrict__ p, size_t n) {
  size_t i = (size_t)blockIdx.x * blockDim.x + threadIdx.x;
  size_t st = (size_t)gridDim.x * blockDim.x;
  for (; i < n; i += st) p[i] = 0.0f;
}

__global__ void k_deg(const int* __restrict__ row, int E, float* __restrict__ deg) {
  int e = blockIdx.x * blockDim.x + threadIdx.x;
  if (e < E) atomicAdd(&deg[row[e]], 1.0f);
}

__global__ void k_dis(float* __restrict__ deg, int n) {
  int i = blockIdx.x * blockDim.x + threadIdx.x;
  if (i < n) {
    float d = deg[i];
    deg[i] = (d > 0.0f) ? rsqrtf(d) : 0.0f;  // d>=1 when >0
  }
}

__global__ void k_norm(const int* __restrict__ row, const int* __restrict__ col,
                       const float* __restrict__ dis, float* __restrict__ nrm, int E) {
  int e = blockIdx.x * blockDim.x + threadIdx.x;
  if (e < E) nrm[e] = dis[row[e]] * dis[col[e]];
}

// scatter: x_out[row] += norm * x_in[col]; thread = (edge, 4-dim chunk)
__global__ void k_prop(const int* __restrict__ row, const int* __restrict__ col,
                       const float* __restrict__ nrm, const float* __restrict__ xin,
                       float* __restrict__ xout, int E) {
  int tid = blockIdx.x * blockDim.x + threadIdx.x;
  if (tid >= E * 16) return;
  int e = tid >> 4;
  int q = (tid & 15) << 2;           // dim offset 0..60 step 4
  int r = row[e], c = col[e];
  float nm = nrm[e];
  const float4 xv = *reinterpret_cast<const float4*>(xin + (size_t)c * NDIM + q);
  float* dst = xout + (size_t)r * NDIM + q;
  atomicAdd(dst + 0, nm * xv.x);
  atomicAdd(dst + 1, nm * xv.y);
  atomicAdd(dst + 2, nm * xv.z);
  atomicAdd(dst + 3, nm * xv.w);
}

__global__ void k_zp(const float* __restrict__ E, const float* x1,
                     const float* __restrict__ x2, float* zp, size_t n) {
  size_t i = (size_t)blockIdx.x * blockDim.x + threadIdx.x;
  size_t st = (size_t)gridDim.x * blockDim.x;
  const float inv3 = 1.0f / 3.0f;
  for (; i < n; i += st) zp[i] = (E[i] + x1[i] + x2[i]) * inv3;  // zp may alias x1
}

// ------------------------------------------------- fused 2-layer MLP (WMMA f32)
// zn = relu(relu(E2 @ W0^T + b0) @ W1^T + b1), each wave owns a 16-row strip.
__global__ __launch_bounds__(128)
void k_mlp(const float* __restrict__ E2, const float* __restrict__ W0,
           const float* __restrict__ b0, const float* __restrict__ W1,
           const float* __restrict__ b1, float* __restrict__ zn) {
  __shared__ float lw0[64 * LW];
  __shared__ float lw1[64 * LW];
  __shared__ float lb0[64];
  __shared__ float lb1[64];
  __shared__ float hls[4][16 * LW];

  int t = threadIdx.x;
  for (int i = t; i < 64 * 64; i += 128) {
    lw0[(i >> 6) * LW + (i & 63)] = W0[i];
    lw1[(i >> 6) * LW + (i & 63)] = W1[i];
  }
  if (t < 64) { lb0[t] = b0[t]; lb1[t] = b1[t]; }
  __syncthreads();

  const int wave = t >> 5, lane = t & 31;
  const int ln = lane & 15, half = lane >> 4;
  const int rowbase = (blockIdx.x * 4 + wave) * 16;
  if (rowbase >= NNODES) return;               // wave-uniform: EXEC stays all-1s

  // ---- layer 0: h = relu(E2 @ W0^T + b0)
  v8f acc0[4] = {};
  #pragma unroll
  for (int kt = 0; kt < 16; ++kt) {
    const int kb = kt * 4 + 2 * half;
    const v2f a = *reinterpret_cast<const v2f*>(E2 + (size_t)(rowbase + ln) * NDIM + kb);
    #pragma unroll
    for (int jt = 0; jt < 4; ++jt) {
      v2f b = { lw0[(jt * 16 + ln) * LW + kb], lw0[(jt * 16 + ln) * LW + kb + 1] };
      acc0[jt] = __builtin_amdgcn_wmma_f32_16x16x4_f32(false, a, false, b,
                                                       (short)0, acc0[jt], false, false);
    }
  }
  float* h = hls[wave];
  #pragma unroll
  for (int jt = 0; jt < 4; ++jt) {
    const float bias = lb0[jt * 16 + ln];
    #pragma unroll
    for (int r = 0; r < 8; ++r) {
      float hv = acc0[jt][r] + bias;
      h[(r + 8 * half) * LW + jt * 16 + ln] = hv > 0.0f ? hv : 0.0f;
    }
  }
  // ---- layer 1: zn = relu(h @ W1^T + b1)  (A fragments re-read from LDS)
  v8f acc1[4] = {};
  #pragma unroll
  for (int kt = 0; kt < 16; ++kt) {
    const int kb = kt * 4 + 2 * half;
    v2f a = { h[ln * LW + kb], h[ln * LW + kb + 1] };
    #pragma unroll
    for (int jt = 0; jt < 4; ++jt) {
      v2f b = { lw1[(jt * 16 + ln) * LW + kb], lw1[(jt * 16 + ln) * LW + kb + 1] };
      acc1[jt] = __builtin_amdgcn_wmma_f32_16x16x4_f32(false, a, false, b,
                                                       (short)0, acc1[jt], false, false);
    }
  }
  #pragma unroll
  for (int jt = 0; jt < 4; ++jt) {
    const float bias = lb1[jt * 16 + ln];
    #pragma unroll
    for (int r = 0; r < 8; ++r) {
      float o = acc1[jt][r] + bias;
      zn[(size_t)(rowbase + r + 8 * half) * NDIM + jt * 16 + ln] = o > 0.0f ? o : 0.0f;
    }
  }
}

// ---------------------- attention scores + softmax fusion (WMMA f32, fused Z)
__global__ __launch_bounds__(128)
void k_attn(const float* __restrict__ zp, const float* __restrict__ zn,
            const float* __restrict__ AW, const float* __restrict__ ab,
            const float* __restrict__ qw, float* __restrict__ Z) {
  __shared__ float law[64 * LW];
  __shared__ float lab[64];
  __shared__ float lq[64];
  __shared__ float sc[4][2][16];

  int t = threadIdx.x;
  for (int i = t; i < 64 * 64; i += 128) law[(i >> 6) * LW + (i & 63)] = AW[i];
  if (t < 64) { lab[t] = ab[t]; lq[t] = qw[t]; }
  __syncthreads();

  const int wave = t >> 5, lane = t & 31;
  const int ln = lane & 15, half = lane >> 4;
  const int rowbase = (blockIdx.x * 4 + wave) * 16;
  if (rowbase >= NNODES) return;

  #pragma unroll
  for (int pass = 0; pass < 2; ++pass) {
    const float* X = pass ? zn : zp;
    v8f acc[4] = {};
    #pragma unroll
    for (int kt = 0; kt < 16; ++kt) {
      const int kb = kt * 4 + 2 * half;
      const v2f a = *reinterpret_cast<const v2f*>(X + (size_t)(rowbase + ln) * NDIM + kb);
      #pragma unroll
      for (int jt = 0; jt < 4; ++jt) {
        v2f b = { law[(jt * 16 + ln) * LW + kb], law[(jt * 16 + ln) * LW + kb + 1] };
        acc[jt] = __builtin_amdgcn_wmma_f32_16x16x4_f32(false, a, false, b,
                                                        (short)0, acc[jt], false, false);
      }
    }
    float part[8];
    #pragma unroll
    for (int r = 0; r < 8; ++r) part[r] = 0.0f;
    #pragma unroll
    for (int jt = 0; jt < 4; ++jt) {
      const float bias = lab[jt * 16 + ln];
      const float qv   = lq[jt * 16 + ln];
      #pragma unroll
      for (int r = 0; r < 8; ++r) part[r] += tanhf(acc[jt][r] + bias) * qv;
    }
    #pragma unroll
    for (int r = 0; r < 8; ++r) {           // reduce over the 16 N-lanes (keeps half bit)
      float s = part[r];
      s += __shfl_xor(s, 1, 32);
      s += __shfl_xor(s, 2, 32);
      s += __shfl_xor(s, 4, 32);
      s += __shfl_xor(s, 8, 32);
      part[r] = s;
    }
    if (ln == 0) {
      #pragma unroll
      for (int r = 0; r < 8; ++r) sc[wave][pass][r + 8 * half] = part[r];
    }
  }
  // softmax over {w_p, w_n} and Z = a0*z_p + a1*z_n   (same-wave LDS: in-order)
  for (int i = lane; i < 16 * NDIM; i += 32) {
    const int row = i >> 6, c = i & 63;
    const float wp = sc[wave][0][row], wn = sc[wave][1][row];
    const float m  = fmaxf(wp, wn);
    const float e0 = __expf(wp - m), e1 = __expf(wn - m);
    const float inv = 1.0f / (e0 + e1);
    const size_t g = (size_t)(rowbase + row) * NDIM + c;
    Z[g] = (e0 * zp[g] + e1 * zn[g]) * inv;
  }
}

// ------------------------------------------------- BPR loss, one wave per b
__device__ __forceinline__ float wave_sum(float v) {
  v += __shfl_xor(v, 1, 32);
  v += __shfl_xor(v, 2, 32);
  v += __shfl_xor(v, 4, 32);
  v += __shfl_xor(v, 8, 32);
  v += __shfl_xor(v, 16, 32);
  return v;
}

__global__ __launch_bounds__(256)
void k_bpr(const float* __restrict__ Z, const int* __restrict__ u,
           const int* __restrict__ v, const int* __restrict__ n,
           const float* __restrict__ w, float* __restrict__ sbpr,
           float* __restrict__ regp) {
  const int wave = threadIdx.x >> 5, lane = threadIdx.x & 31;
  const int b = blockIdx.x * 8 + wave;
  if (b >= NB) return;

  const int d = lane * 2;
  const float2 uu = *reinterpret_cast<const float2*>(Z + (size_t)u[b] * NDIM + d);
  const float2 vv = *reinterpret_cast<const float2*>(Z + (size_t)v[b] * NDIM + d);

  float reg = uu.x * uu.x + uu.y * uu.y + vv.x * vv.x + vv.y * vv.y;
  const float pos = wave_sum(uu.x * vv.x + uu.y * vv.y);

  const float wb  = w[b];
  const float sg  = (wb > 0.0f) ? 1.0f : ((wb < 0.0f) ? -1.0f : 0.0f);
  const float sp  = (-0.5f * sg + 1.5f) * pos;

  float acc = 0.0f;
  for (int k = 0; k < NK; ++k) {
    const float2 nn = *reinterpret_cast<const float2*>(Z + (size_t)n[b * NK + k] * NDIM + d);
    reg += nn.x * nn.x + nn.y * nn.y;
    const float neg = wave_sum(uu.x * nn.x + uu.y * nn.y);
    const float x = sp - neg;
    acc += fminf(x, 0.0f) - log1pf(expf(-fabsf(x)));   // stable log_sigmoid
  }
  reg = wave_sum(reg);
  if (lane == 0) { sbpr[b] = acc; regp[b] = reg; }
}

__global__ __launch_bounds__(256)
void k_final(const float* __restrict__ sbpr, const float* __restrict__ regp,
             float* __restrict__ out) {
  __shared__ float red[256];
  float s = 0.0f;
  for (int i = threadIdx.x; i < NB; i += 256) s += -sbpr[i] + REGC * regp[i];
  red[threadIdx.x] = s;
  __syncthreads();
  for (int o = 128; o > 0; o >>= 1) {
    if (threadIdx.x < o) red[threadIdx.x] += red[threadIdx.x + o];
    __syncthreads();
  }
  if (threadIdx.x == 0) out[0] = red[0];
}

// ---------------------------------------------------------------- launcher
extern "C" void kernel_launch(void* const* d_in, const int* in_sizes, int n_in,
                              void* d_out, int out_size, void* d_ws, size_t ws_size,
                              hipStream_t stream) {
  const float* E   = (const float*)d_in[0];
  const float* E2  = (const float*)d_in[1];
  const float* W0  = (const float*)d_in[2];
  const float* b0  = (const float*)d_in[3];
  const float* W1  = (const float*)d_in[4];
  const float* b1  = (const float*)d_in[5];
  const float* AW  = (const float*)d_in[6];
  const float* ab  = (const float*)d_in[7];
  const float* qw  = (const float*)d_in[8];
  const float* w   = (const float*)d_in[9];
  const int*   edge= (const int*)d_in[10];
  const int*   u   = (const int*)d_in[11];
  const int*   v   = (const int*)d_in[12];
  const int*   nn  = (const int*)d_in[13];

  const int Etot = in_sizes[10] / 2;           // 2,000,000 directed edges
  const int* row = edge;
  const int* col = edge + Etot;

  float* ws = (float*)d_ws;
  size_t off = 0;
  float* dis = ws + off; off += 150016;                 // deg -> deg_inv_sqrt (in place)
  float* nrm = ws + off; off += (size_t)Etot;           // per-edge norm
  float* x1  = ws + off; off += (size_t)NNODES * NDIM;  // layer1 -> z_p
  float* x2  = ws + off; off += (size_t)NNODES * NDIM;  // layer2 -> Z
  float* zn  = ws + off; off += (size_t)NNODES * NDIM;  // MLP branch
  float* sb  = ws + off; off += NB;
  float* rg  = ws + off; off += NB;

  const size_t nfeat = (size_t)NNODES * NDIM;
  k_zero<<<1024, 256, 0, stream>>>(dis, (size_t)150016);
  k_zero<<<4096, 256, 0, stream>>>(x1, nfeat);
  k_zero<<<4096, 256, 0, stream>>>(x2, nfeat);

  k_deg <<<(Etot + 255) / 256, 256, 0, stream>>>(row, Etot, dis);
  k_dis <<<(NNODES + 255) / 256, 256, 0, stream>>>(dis, NNODES);
  k_norm<<<(Etot + 255) / 256, 256, 0, stream>>>(row, col, dis, nrm, Etot);

  const int pthreads = Etot * 16;
  const int pblocks  = (pthreads + 255) / 256;
  k_prop<<<pblocks, 256, 0, stream>>>(row, col, nrm, E,  x1, Etot);
  k_prop<<<pblocks, 256, 0, stream>>>(row, col, nrm, x1, x2, Etot);

  k_zp<<<4096, 256, 0, stream>>>(E, x1, x2, x1, nfeat);  // z_p -> x1

  const int waves   = NNODES / 16;                       // 9375 exactly
  const int gblocks = (waves + 3) / 4;                   // 4 waves / block
  k_mlp <<<gblocks, 128, 0, stream>>>(E2, W0, b0, W1, b1, zn);
  k_attn<<<gblocks, 128, 0, stream>>>(x1, zn, AW, ab, qw, x2);  // Z -> x2

  k_bpr  <<<NB / 8, 256, 0, stream>>>(x2, u, v, nn, w, sb, rg);
  k_final<<<1, 256, 0, stream>>>(sb, rg, (float*)d_out);
}